// GAT_57939108823675
// MI455X (gfx1250) — compile-verified
//
#include <hip/hip_runtime.h>
#include <math.h>

// ---------- CDNA5 WMMA vector types ----------
typedef __attribute__((ext_vector_type(2))) float v2f;
typedef __attribute__((ext_vector_type(8))) float v8f;

#define NEG_SLOPE 0.2f

__device__ __forceinline__ float leaky(float x) { return x > 0.f ? x : NEG_SLOPE * x; }

// Float atomic max via monotone int bit-mapping (lowers to native
// GLOBAL_ATOMIC_MAX_I32 / GLOBAL_ATOMIC_MIN_U32; init value must be -inf).
__device__ __forceinline__ void atomicMaxF(float* addr, float v) {
  if (v >= 0.f) atomicMax((int*)addr, __float_as_int(v));
  else          atomicMin((unsigned int*)addr, __float_as_uint(v));
}

// ---------- utility: fill a float buffer ----------
__global__ void gat_fill(float* __restrict__ p, float v, int n) {
  int i = blockIdx.x * blockDim.x + threadIdx.x;
  if (i < n) p[i] = v;
}

// ---------- FP32 WMMA GEMM: C[M,Nc] = A[M,K] @ B[K,Nc] ----------
// One wave per (16 x 16*TNT) output strip: TNT accumulators share one A
// operand per K-step. B row pointers strength-reduced (+= 4*Nc per step) and
// the loop is software-pipelined one K-step so loads overlap the 4 WMMAs.
// K multiple of 4; M multiple of 16 (50000 = 3125*16); Nc masked per column.
template <int TNT>
__global__ __launch_bounds__(256) void gat_wmma_gemm(
    const float* __restrict__ A, const float* __restrict__ B,
    float* __restrict__ C, int M, int K, int Nc) {
  const int lane = threadIdx.x & 31;
  const int wave = threadIdx.x >> 5;
  const int l16  = lane & 15;
  const int hi   = lane >> 4;              // 0: lanes 0-15, 1: lanes 16-31
  const int tilesN  = (Nc + 15) >> 4;
  const int stripsN = (tilesN + TNT - 1) / TNT;
  const int tilesM  = M >> 4;
  const int wid = blockIdx.x * 8 + wave;   // wave-uniform -> EXEC stays full
  if (wid >= tilesM * stripsN) return;
  const int tm  = wid / stripsN;
  const int tnb = (wid % stripsN) * TNT;   // first N-tile of this strip

  const int arow = tm * 16 + l16;          // A row this lane feeds

  int   col[TNT];
  int   colOk[TNT];
  float colMask[TNT];
  const float* __restrict__ b0[TNT];       // row kk   pointer per tile
  const float* __restrict__ b1[TNT];       // row kk+1 pointer per tile
#pragma unroll
  for (int t = 0; t < TNT; ++t) {
    col[t] = (tnb + t) * 16 + l16;
    colOk[t] = (col[t] < Nc);
    colMask[t] = colOk[t] ? 1.0f : 0.0f;
    const float* base = B + (colOk[t] ? col[t] : 0) + (size_t)(hi * 2) * Nc;
    b0[t] = base;
    b1[t] = base + Nc;
  }
  const size_t bstep = (size_t)4 * Nc;     // advance one K-step (4 rows)

  // A pointer with this lane's K phase folded in; step = 4 floats.
  const float* __restrict__ ap = A + (size_t)arow * K + hi * 2;

  v8f acc[TNT];
#pragma unroll
  for (int t = 0; t < TNT; ++t) acc[t] = (v8f){};

  const int nk = K >> 2;                   // number of 16x16x4 K-steps

  // ---- prologue: load step 0 ----
  v2f a_c;
  a_c.x = ap[0];
  a_c.y = ap[1];
  v2f b_c[TNT];
#pragma unroll
  for (int t = 0; t < TNT; ++t) {
    b_c[t].x = b0[t][0] * colMask[t];
    b_c[t].y = b1[t][0] * colMask[t];
  }
  ap += 4;

  // ---- steady state: issue loads for step i+1, then WMMAs for step i ----
  for (int i = 0; i < nk - 1; ++i) {
    const float a0n = ap[0];
    const float a1n = ap[1];
    float bxn[TNT], byn[TNT];
#pragma unroll
    for (int t = 0; t < TNT; ++t) {
      b0[t] += bstep;
      b1[t] += bstep;
      bxn[t] = b0[t][0];
      byn[t] = b1[t][0];
    }
#pragma unroll
    for (int t = 0; t < TNT; ++t) {
      // D = A(16x4) * B(4x16) + C   -> v_wmma_f32_16x16x4_f32
      acc[t] = __builtin_amdgcn_wmma_f32_16x16x4_f32(
          /*neg_a=*/false, a_c, /*neg_b=*/false, b_c[t],
          /*c_mod=*/(short)0, acc[t], /*reuse_a=*/false, /*reuse_b=*/false);
    }
    a_c.x = a0n;
    a_c.y = a1n;
#pragma unroll
    for (int t = 0; t < TNT; ++t) {
      b_c[t].x = bxn[t] * colMask[t];
      b_c[t].y = byn[t] * colMask[t];
    }
    ap += 4;
  }
  // ---- epilogue: last K-step ----
#pragma unroll
  for (int t = 0; t < TNT; ++t) {
    acc[t] = __builtin_amdgcn_wmma_f32_16x16x4_f32(
        false, a_c, false, b_c[t], (short)0, acc[t], false, false);
  }

#pragma unroll
  for (int t = 0; t < TNT; ++t) {
    if (colOk[t]) {
#pragma unroll
      for (int v = 0; v < 8; ++v) {
        const int r = tm * 16 + hi * 8 + v;  // C/D layout: VGPR v -> M=v (+8 hi half)
        C[(size_t)r * Nc + col[t]] = acc[t][v];
      }
    }
  }
}

// ---------- el/er attention dots: el[n,h] = sum_d feat[n,h,d]*al[h,d] ----------
// D is a multiple of 4 and rows are 16B-aligned -> float4 loads.
__global__ void gat_attn_scores(const float* __restrict__ feat,
                                const float* __restrict__ al,
                                const float* __restrict__ ar,
                                float* __restrict__ el, float* __restrict__ er,
                                int N, int H, int D) {
  int i = blockIdx.x * blockDim.x + threadIdx.x;
  if (i >= N * H) return;
  const int h = i % H;
  const float4* f  = (const float4*)(feat + (size_t)i * D);
  const float4* a4 = (const float4*)(al + (size_t)h * D);
  const float4* r4 = (const float4*)(ar + (size_t)h * D);
  float sl = 0.f, sr = 0.f;
  const int q = D >> 2;
  for (int d = 0; d < q; ++d) {
    const float4 v = f[d];
    const float4 la = a4[d];
    const float4 ra = r4[d];
    sl += v.x * la.x + v.y * la.y + v.z * la.z + v.w * la.w;
    sr += v.x * ra.x + v.y * ra.y + v.z * ra.z + v.w * ra.w;
  }
  el[i] = sl;
  er[i] = sr;
}

// ---------- edge softmax pass A: per-dst max ----------
__global__ void gat_edge_max(const int* __restrict__ src, const int* __restrict__ dst,
                             const float* __restrict__ el, const float* __restrict__ er,
                             float* __restrict__ emax, int E, int H) {
  int e = blockIdx.x * blockDim.x + threadIdx.x;
  if (e >= E) return;
  const int s = src[e], d = dst[e];
  for (int h = 0; h < H; ++h) {
    const float x = leaky(el[s * H + h] + er[d * H + h]);
    atomicMaxF(&emax[d * H + h], x);
  }
}

// ---------- edge softmax pass B: exp + per-dst denom ----------
__global__ void gat_edge_exp(const int* __restrict__ src, const int* __restrict__ dst,
                             const float* __restrict__ el, const float* __restrict__ er,
                             const float* __restrict__ emax,
                             float* __restrict__ eexp, float* __restrict__ denom,
                             int E, int H) {
  int e = blockIdx.x * blockDim.x + threadIdx.x;
  if (e >= E) return;
  const int s = src[e], d = dst[e];
  for (int h = 0; h < H; ++h) {
    const float x = leaky(el[s * H + h] + er[d * H + h]);
    const float ex = expf(x - emax[d * H + h]);
    eexp[(size_t)e * H + h] = ex;
    atomicAdd(&denom[d * H + h], ex);
  }
}

// ---------- message aggregation: rst[dst] += feat[src] * alpha ----------
// Thread handles 4 contiguous floats (float4 gather + 4 L2-resident atomics).
// F = H*D; 4 | D so a chunk never crosses a head boundary. feat (51MB) and
// rst (51MB) both fit the 192MB L2 simultaneously -> atomics run at L2.
__global__ void gat_edge_aggregate(const int* __restrict__ src, const int* __restrict__ dst,
                                   const float* __restrict__ feat,
                                   const float* __restrict__ eexp,
                                   const float* __restrict__ denom,
                                   float* __restrict__ rst,
                                   int E, int H, int D) {
  const int F = H * D;
  const int chunks = F >> 2;
  long long idx = (long long)blockIdx.x * blockDim.x + threadIdx.x;
  if (idx >= (long long)E * chunks) return;
  const int e = (int)(idx / chunks);
  const int c = (int)(idx % chunks);
  const int h = (c * 4) / D;
  const int s = src[e], d = dst[e];
  const float alpha = eexp[(size_t)e * H + h] / denom[(size_t)d * H + h];
  const float4 f = *(const float4*)(feat + (size_t)s * F + c * 4);
  float* out = rst + (size_t)d * F + c * 4;
  atomicAdd(out + 0, f.x * alpha);
  atomicAdd(out + 1, f.y * alpha);
  atomicAdd(out + 2, f.z * alpha);
  atomicAdd(out + 3, f.w * alpha);
}

// ---------- bias + ELU (layer-1 activation), in place ----------
__global__ void gat_bias_elu(float* __restrict__ x, const float* __restrict__ bias,
                             int N, int F) {
  int i = blockIdx.x * blockDim.x + threadIdx.x;
  if (i >= N * F) return;
  const float v = x[i] + bias[i % F];
  x[i] = v > 0.f ? v : (expf(v) - 1.f);
}

// ---------- final: bias + log_softmax over 40 classes, in place ----------
__global__ void gat_bias_logsoftmax40(float* __restrict__ out,
                                      const float* __restrict__ bias, int N) {
  int n = blockIdx.x * blockDim.x + threadIdx.x;
  if (n >= N) return;
  float* row = out + (size_t)n * 40;
  float v[40];
  float m = -__builtin_huge_valf();
#pragma unroll
  for (int j = 0; j < 40; ++j) {
    v[j] = row[j] + bias[j];
    m = v[j] > m ? v[j] : m;
  }
  float ssum = 0.f;
#pragma unroll
  for (int j = 0; j < 40; ++j) ssum += expf(v[j] - m);
  const float lse = m + logf(ssum);
#pragma unroll
  for (int j = 0; j < 40; ++j) row[j] = v[j] - lse;
}

// =====================================================================
extern "C" void kernel_launch(void* const* d_in, const int* in_sizes, int n_in,
                              void* d_out, int out_size, void* d_ws, size_t ws_size,
                              hipStream_t stream) {
  const float* features = (const float*)d_in[0];
  const int*   src      = (const int*)d_in[1];
  const int*   dst      = (const int*)d_in[2];
  const float* W1       = (const float*)d_in[3];
  const float* al1      = (const float*)d_in[4];
  const float* ar1      = (const float*)d_in[5];
  const float* b1       = (const float*)d_in[6];
  const float* W2       = (const float*)d_in[7];
  const float* al2      = (const float*)d_in[8];
  const float* ar2      = (const float*)d_in[9];
  const float* b2       = (const float*)d_in[10];
  float* out            = (float*)d_out;

  const int IN = 256, H1 = 4, D1 = 64, F1 = 256, OUTC = 40;
  const int N = in_sizes[0] / IN;
  const int E = in_sizes[1];

  // ---- workspace layout (floats) ----
  float* ws    = (float*)d_ws;
  float* feat1 = ws;                               // N*256
  float* rst1  = feat1 + (size_t)N * F1;           // N*256 (becomes h after ELU)
  float* el1   = rst1 + (size_t)N * F1;            // N*4
  float* er1   = el1 + (size_t)N * H1;             // N*4
  float* max1  = er1 + (size_t)N * H1;             // N*4
  float* den1  = max1 + (size_t)N * H1;            // N*4
  float* eexp1 = den1 + (size_t)N * H1;            // E*4
  float* feat2 = eexp1 + (size_t)E * H1;           // N*40
  float* el2   = feat2 + (size_t)N * OUTC;         // N
  float* er2   = el2 + (size_t)N;                  // N
  float* max2  = er2 + (size_t)N;                  // N
  float* den2  = max2 + (size_t)N;                 // N
  float* eexp2 = den2 + (size_t)N;                 // E

  const int TB = 256;
  const float NEG_INF = -__builtin_huge_valf();
  auto cdiv = [](long long a, long long b) { return (unsigned)((a + b - 1) / b); };

  // ===================== Layer 1 =====================
  gat_fill<<<cdiv((long long)N * F1, TB), TB, 0, stream>>>(rst1, 0.f, N * F1);
  gat_fill<<<cdiv((long long)N * H1, TB), TB, 0, stream>>>(max1, NEG_INF, N * H1);
  gat_fill<<<cdiv((long long)N * H1, TB), TB, 0, stream>>>(den1, 0.f, N * H1);

  {  // feat1 = X @ W1  (M=N, K=256, Nc=256): 4 N-tiles per wave
    const int waves = (N / 16) * (((F1 + 15) / 16 + 3) / 4);
    gat_wmma_gemm<4><<<cdiv(waves, 8), 256, 0, stream>>>(features, W1, feat1, N, IN, F1);
  }
  gat_attn_scores<<<cdiv((long long)N * H1, TB), TB, 0, stream>>>(feat1, al1, ar1, el1, er1, N, H1, D1);
  gat_edge_max<<<cdiv(E, TB), TB, 0, stream>>>(src, dst, el1, er1, max1, E, H1);
  gat_edge_exp<<<cdiv(E, TB), TB, 0, stream>>>(src, dst, el1, er1, max1, eexp1, den1, E, H1);
  gat_edge_aggregate<<<cdiv((long long)E * (F1 / 4), TB), TB, 0, stream>>>(
      src, dst, feat1, eexp1, den1, rst1, E, H1, D1);
  gat_bias_elu<<<cdiv((long long)N * F1, TB), TB, 0, stream>>>(rst1, b1, N, F1);

  // ===================== Layer 2 =====================
  gat_fill<<<cdiv((long long)N * OUTC, TB), TB, 0, stream>>>(out, 0.f, N * OUTC);
  gat_fill<<<cdiv(N, TB), TB, 0, stream>>>(max2, NEG_INF, N);
  gat_fill<<<cdiv(N, TB), TB, 0, stream>>>(den2, 0.f, N);

  {  // feat2 = h @ W2  (M=N, K=256, Nc=40 -> 3 live tiles, 4th masked)
    const int waves = (N / 16) * (((OUTC + 15) / 16 + 3) / 4);
    gat_wmma_gemm<4><<<cdiv(waves, 8), 256, 0, stream>>>(rst1, W2, feat2, N, F1, OUTC);
  }
  gat_attn_scores<<<cdiv(N, TB), TB, 0, stream>>>(feat2, al2, ar2, el2, er2, N, 1, OUTC);
  gat_edge_max<<<cdiv(E, TB), TB, 0, stream>>>(src, dst, el2, er2, max2, E, 1);
  gat_edge_exp<<<cdiv(E, TB), TB, 0, stream>>>(src, dst, el2, er2, max2, eexp2, den2, E, 1);
  gat_edge_aggregate<<<cdiv((long long)E * (OUTC / 4), TB), TB, 0, stream>>>(
      src, dst, feat2, eexp2, den2, out, E, 1, OUTC);
  gat_bias_logsoftmax40<<<cdiv(N, TB), TB, 0, stream>>>(out, b2, N);
}